// ParallelLinearTransposed_77627238908438
// MI455X (gfx1250) — compile-verified
//
#include <hip/hip_runtime.h>

// Per-edge tiny linear: out[e] = x[e] @ W[e]^T + b[e]
// E=50000, B=512, N=2, O=2  -> memory-bound (~410 MB traffic, ~17.6us @ 23.3 TB/s).
// Compute routed through V_WMMA_F32_16X16X4_F32: two K=2 edges packed
// block-diagonally into K=4; bias folded into the C accumulator.

typedef float v2f __attribute__((ext_vector_type(2)));
typedef float v8f __attribute__((ext_vector_type(8)));

#define E_CONST 50000
#define B_CONST 512

__global__ __launch_bounds__(256)
void ParallelLinearTransposed_77627238908438_kernel(
    const float* __restrict__ x,   // (E, B, 2)
    const float* __restrict__ W,   // (E, 2, 2)
    const float* __restrict__ b,   // (E, 2)
    float* __restrict__ out)       // (E, B, 2)
{
    const int e0    = blockIdx.x * 2;          // this block handles edges e0, e0+1
    const int tid   = threadIdx.x;
    const int lane  = tid & 31;                // wave32 lane
    const int wave  = tid >> 5;                // 0..7
    const int half  = lane >> 4;               // 0 -> edge e0, 1 -> edge e0+1
    const int rowin = lane & 15;               // batch row within 16-row tile

    // ---- Uniform per-block parameters (scalar loads: W is 8 floats, b is 4) ----
    const float* We = W + (size_t)e0 * 4;      // W[e,o,n] flat = e*4 + o*2 + n
    const float we00 = We[0], we01 = We[1], we10 = We[2], we11 = We[3];
    const float wp00 = We[4], wp01 = We[5], wp10 = We[6], wp11 = We[7];
    const float* Be = b + (size_t)e0 * 2;
    const float be0 = Be[0], be1 = Be[1], bp0 = Be[2], bp1 = Be[3];

    // ---- A matrix (16x4 f32), block-diagonal in K ----
    // ISA layout: lanes 0-15 hold (K0,K1) in VGPR0/1 for row M=lane;
    //             lanes 16-31 hold (K2,K3) for row M=lane-16.
    // A[0,0:2]=W[e0] row0, A[1,0:2]=W[e0] row1, A[2,2:4]=W[e0+1] row0, A[3,2:4]=W[e0+1] row1.
    v2f a; a.x = 0.0f; a.y = 0.0f;
    if      (lane == 0)  { a.x = we00; a.y = we01; }
    else if (lane == 1)  { a.x = we10; a.y = we11; }
    else if (lane == 18) { a.x = wp00; a.y = wp01; }
    else if (lane == 19) { a.x = wp10; a.y = wp11; }

    // ---- C matrix = bias (rows 0..3), zero elsewhere ----
    // D/C layout: VGPR r holds row M=r in lanes 0-15 (rows 8..15 in lanes 16-31).
    v8f c = {0.f,0.f,0.f,0.f,0.f,0.f,0.f,0.f};
    if (lane < 16) { c[0] = be0; c[1] = be1; c[2] = bp0; c[3] = bp1; }

    // Same flat index serves x and out (both (.,B,2)):
    // lanes 0-15 stream edge e0 rows r0..r0+15, lanes 16-31 stream edge e0+1.
    const size_t ebase = (size_t)(e0 + half) * B_CONST;

#pragma unroll
    for (int i = 0; i < 4; ++i) {
        const int    r0  = wave * 64 + i * 16;
        const size_t idx = (ebase + (size_t)(r0 + rowin)) * 2;

        // ---- B matrix (4x16 f32): K rows {0,1}=x[e0]^T, {2,3}=x[e0+1]^T ----
        // ISA layout: VGPR0 = K0 (lanes 0-15) / K2 (lanes 16-31); VGPR1 = K1/K3.
        // Each lane's contiguous float2 of x lands exactly in (VGPR0, VGPR1).
        v2f bm = __builtin_nontemporal_load((const v2f*)(x + idx));

        // D = A x B + C  (rows 0,1 -> out[e0], rows 2,3 -> out[e0+1], cols = batch)
        v8f d = __builtin_amdgcn_wmma_f32_16x16x4_f32(
            /*neg_a=*/false, a, /*neg_b=*/false, bm,
            /*c_mod=*/(short)0, c, /*reuse_a=*/false, /*reuse_b=*/false);

        // Move edge e0+1 results (rows 2,3, live in lanes 0-15) to lanes 16-31.
        const float s2 = __shfl_xor(d[2], 16, 32);
        const float s3 = __shfl_xor(d[3], 16, 32);

        v2f ov;
        ov.x = half ? s2 : d[0];
        ov.y = half ? s3 : d[1];
        __builtin_nontemporal_store(ov, (v2f*)(out + idx));
    }
}

extern "C" void kernel_launch(void* const* d_in, const int* in_sizes, int n_in,
                              void* d_out, int out_size, void* d_ws, size_t ws_size,
                              hipStream_t stream) {
    (void)in_sizes; (void)n_in; (void)d_ws; (void)ws_size; (void)out_size;
    const float* x = (const float*)d_in[0];
    const float* W = (const float*)d_in[1];
    const float* b = (const float*)d_in[2];
    float* out = (float*)d_out;

    dim3 grid(E_CONST / 2);   // 25000 blocks, one edge pair each
    dim3 block(256);          // 8 wave32s; each wave: 4 WMMAs covering 64 batch rows
    hipLaunchKernelGGL(ParallelLinearTransposed_77627238908438_kernel,
                       grid, block, 0, stream, x, W, b, out);
}